// HeteroSTBlock_59785944760479
// MI455X (gfx1250) — compile-verified
//
#include <hip/hip_runtime.h>
#include <math.h>

typedef __attribute__((ext_vector_type(2))) float v2f;
typedef __attribute__((ext_vector_type(8))) float v8f;

#define B_    16
#define T_    20
#define TM_   18
#define NR_   64
#define ND_   256
#define NP_   1024
#define CMID_ 128
#define COUT_ 64

// ---------------- pack: [B,C,T,N] -> [(b*N+n)*TMk + t, C] row-major ----------------
__global__ void pack_nchw(const float* __restrict__ in, float* __restrict__ out,
                          int C, int T, int N, int TMk) {
  long long idx = (long long)blockIdx.x * blockDim.x + threadIdx.x;
  long long total = (long long)B_ * N * TMk * C;
  if (idx >= total) return;
  int c = (int)(idx % C);
  int t = (int)((idx / C) % TMk);
  long long nn = idx / ((long long)C * TMk);
  int n = (int)(nn % N);
  int b = (int)(nn / N);
  out[idx] = in[(((long long)b * C + c) * T + t) * N + n];
}

// ---------------- weight repack: w[O,C,3] -> W[O, 3*C] with col = dt*C + c ----------------
__global__ void repack_w3(const float* __restrict__ w, float* __restrict__ out, int O, int C) {
  int idx = blockIdx.x * blockDim.x + threadIdx.x;
  int total = O * 3 * C;
  if (idx >= total) return;
  int c  = idx % C;
  int dt = (idx / C) % 3;
  int o  = idx / (3 * C);
  out[idx] = w[(o * C + c) * 3 + dt];
}

// ---------------- generic fp32 WMMA GEMM: D[M,N] = Aview[M,K] * W[N,K]^T + bias ----------------
// A row m lives at A + (m/rpn)*cs + (m%rpn)*rs (lets conv windows alias flat buffers).
// Block = 8 waves, one 64-column N-group per block. The W panel (64 x K, padded rows)
// is DMA'd into LDS with global_load_async_to_lds_b32 (ASYNCcnt) and shared by all
// 8 waves; each wave computes a 16(M) x 64(N) strip with V_WMMA_F32_16X16X4_F32,
// B-fragments from LDS (ds_load_b64), A-fragments prefetched + pipelined from global.
__global__ __launch_bounds__(256) void gemm_wmma_f32(
    const float* __restrict__ A, int rpn, long long cs, long long rs,
    int M, int K,
    const float* __restrict__ W, const float* __restrict__ bias,
    float* __restrict__ D, int N)
{
  extern __shared__ float shW[];             // (K+2) * 64 floats
  const int groupsN = N >> 6;
  const int tilesM  = M >> 4;
  const int bm = blockIdx.x / groupsN;
  const int ng = blockIdx.x - bm * groupsN;
  const int Kp = K + 2;

  // ---- async-DMA the 64 x K weight panel into LDS (ASYNCcnt-tracked) ----
  {
    const float* wsrc = W + ((long long)ng << 6) * K;
    // ISA 10.2: flat LDS address low 32 bits == LDS byte offset
    unsigned shbase = (unsigned)(unsigned long long)(void*)shW;
    int totalW = K << 6;                     // 64*K, always a multiple of 256 here
    for (int i = threadIdx.x; i < totalW; i += 256) {
      int r = i / K;
      int c = i - r * K;
      const float* g = wsrc + i;
      unsigned loff = shbase + (unsigned)(r * Kp + c) * 4u;
      asm volatile("global_load_async_to_lds_b32 %0, %1, off"
                   :: "v"(loff), "v"(g) : "memory");
    }
    asm volatile("s_wait_asynccnt 0x0" ::: "memory");
    __syncthreads();
  }

  const int wid = threadIdx.x >> 5;
  const int mi = (bm << 3) + wid;
  if (mi >= tilesM) return;                  // tail waves still helped stage W

  const int lane = threadIdx.x & 31;
  const int half = lane >> 4;
  const int lr   = lane & 15;

  const int m = (mi << 4) + lr;
  const float* arow = A + (long long)(m / rpn) * cs + (long long)(m % rpn) * rs;
  __builtin_prefetch(arow, 0, 3);            // global_prefetch_b8 for the A row

  const float* wl0 = shW + lr * Kp;
  const float* wl1 = shW + (lr + 16) * Kp;
  const float* wl2 = shW + (lr + 32) * Kp;
  const float* wl3 = shW + (lr + 48) * Kp;

  const int kofs = half << 1;
  v8f c0 = {}; v8f c1 = {}; v8f c2 = {}; v8f c3 = {};
  v2f a = *(const v2f*)(arow + kofs);        // pipelined A fragment
  for (int k = 0; k < K; k += 4) {
    v2f a_cur = a;
    if (k + 4 < K) a = *(const v2f*)(arow + k + 4 + kofs);   // next A while WMMA runs
    int kk = k + kofs;
    v2f b0 = *(const v2f*)(wl0 + kk);
    v2f b1 = *(const v2f*)(wl1 + kk);
    v2f b2 = *(const v2f*)(wl2 + kk);
    v2f b3 = *(const v2f*)(wl3 + kk);
    c0 = __builtin_amdgcn_wmma_f32_16x16x4_f32(false, a_cur, false, b0, (short)0, c0, false, false);
    c1 = __builtin_amdgcn_wmma_f32_16x16x4_f32(false, a_cur, false, b1, (short)0, c1, false, false);
    c2 = __builtin_amdgcn_wmma_f32_16x16x4_f32(false, a_cur, false, b2, (short)0, c2, false, false);
    c3 = __builtin_amdgcn_wmma_f32_16x16x4_f32(false, a_cur, false, b3, (short)0, c3, false, false);
  }

  const int n0 = (ng << 6) + lr;
  float bv0 = 0.f, bv1 = 0.f, bv2 = 0.f, bv3 = 0.f;
  if (bias) { bv0 = bias[n0]; bv1 = bias[n0 + 16]; bv2 = bias[n0 + 32]; bv3 = bias[n0 + 48]; }
#pragma unroll
  for (int j = 0; j < 8; ++j) {
    int mrow = (mi << 4) + j + (half << 3);
    float* drow = D + (long long)mrow * N;
    drow[n0]      = c0[j] + bv0;
    drow[n0 + 16] = c1[j] + bv1;
    drow[n0 + 32] = c2[j] + bv2;
    drow[n0 + 48] = c3[j] + bv3;
  }
}

// ---------------- stage-1 property GLU: fp = (y[:,:128] + pad(x)[t+2]) * sigmoid(y[:,128:]) ----------------
__global__ void glu_prop1(const float* __restrict__ y, const float* __restrict__ Xp,
                          float* __restrict__ fp) {
  long long idx = (long long)blockIdx.x * blockDim.x + threadIdx.x;
  long long total = (long long)B_ * NP_ * TM_ * CMID_;
  if (idx >= total) return;
  int o = (int)(idx & 127);
  long long m = idx >> 7;
  int t = (int)(m % TM_);
  long long node = m / TM_;
  float p = y[m * 256 + o];
  float q = y[m * 256 + 128 + o];
  float xa = (o < 32) ? Xp[node * (T_ * 32) + (long long)(t + 2) * 32 + o] : 0.f;
  fp[idx] = (p + xa) * (1.f / (1.f + __expf(-q)));
}

// ---------------- tiny temporal conv + GLU for outside/time branches (N=1) ----------------
__global__ void tconv16_glu(const float* __restrict__ x, const float* __restrict__ w,
                            const float* __restrict__ bias, float* __restrict__ outFlat) {
  int idx = blockIdx.x * blockDim.x + threadIdx.x;
  int total = B_ * TM_ * CMID_;
  if (idx >= total) return;
  int o = idx & 127;
  int t = (idx >> 7) % TM_;
  int b = idx / (CMID_ * TM_);
  float y1 = bias[o], y2 = bias[o + 128];
  for (int c = 0; c < 16; ++c) {
    const float* xc = x + (b * 16 + c) * T_;
#pragma unroll
    for (int dt = 0; dt < 3; ++dt) {
      float xv = xc[t + dt];
      y1 += xv * w[(o * 16 + c) * 3 + dt];
      y2 += xv * w[((o + 128) * 16 + c) * 3 + dt];
    }
  }
  float xa = (o < 16) ? x[(b * 16 + o) * T_ + t + 2] : 0.f;
  outFlat[idx] = (y1 + xa) * (1.f / (1.f + __expf(-y2)));
}

// ---------------- misc elementwise ----------------
__global__ void fill_f32(float* p, long long n, float v) {
  long long i = (long long)blockIdx.x * blockDim.x + threadIdx.x;
  if (i < n) p[i] = v;
}
__global__ void deg_count(const int* __restrict__ idx, int E, float* __restrict__ deg) {
  int e = blockIdx.x * blockDim.x + threadIdx.x;
  if (e < E) atomicAdd(&deg[idx[e]], 1.f);
}
__global__ void to_dinv(float* d, int n) {
  int i = blockIdx.x * blockDim.x + threadIdx.x;
  if (i < n) { float v = d[i]; d[i] = (v > 0.f) ? rsqrtf(v) : 0.f; }
}
__global__ void acc_init(float* __restrict__ acc, long long rows,
                         const float* b0, const float* b1, const float* b2,
                         const float* b3, const float* b4) {
  long long i = (long long)blockIdx.x * blockDim.x + threadIdx.x;
  if (i >= rows * CMID_) return;
  int c = (int)(i & 127);
  float v = b0[c];
  if (b1) v += b1[c];
  if (b2) v += b2[c];
  if (b3) v += b3[c];
  if (b4) v += b4[c];
  acc[i] = v;
}
__global__ void relu_inplace(float* p, long long n) {
  long long i = (long long)blockIdx.x * blockDim.x + threadIdx.x;
  if (i < n) p[i] = fmaxf(p[i], 0.f);
}

// ---------------- GCN scatter (symmetric-normalized segment sum, float atomics) ----------------
__global__ void scatter_edges(const float* __restrict__ H, const int* __restrict__ ei, int E,
                              const float* __restrict__ dis, const float* __restrict__ did,
                              float* __restrict__ acc) {
  int e = blockIdx.x;
  int c = threadIdx.x;
  int src = ei[e], dst = ei[E + e];
  float norm = dis[src] * did[dst];
  const float* h = H + (long long)src * (TM_ * CMID_) + c;
  float* a = acc + (long long)dst * (TM_ * CMID_) + c;
  for (int t = 0; t < TM_; ++t)
    atomicAdd(a + t * CMID_, h[t * CMID_] * norm);
}
__global__ void scatter_self(const float* __restrict__ H, const float* __restrict__ dinv,
                             float* __restrict__ acc) {
  int n = blockIdx.x;
  int c = threadIdx.x;
  float dv = dinv[n];
  float norm = dv * dv;
  const float* h = H + (long long)n * (TM_ * CMID_) + c;
  float* a = acc + (long long)n * (TM_ * CMID_) + c;
  for (int t = 0; t < TM_; ++t)
    atomicAdd(a + t * CMID_, h[t * CMID_] * norm);
}

// ---------------- LayerNorm(64) epilogues writing NCHW output ----------------
__global__ void ln64_out(const float* __restrict__ flat, const float* __restrict__ g,
                         const float* __restrict__ bb, float* __restrict__ out,
                         int rows, int Tdim, int Nn) {
  int m = blockIdx.x * blockDim.x + threadIdx.x;
  if (m >= rows) return;
  const float* r = flat + (long long)m * COUT_;
  float mu = 0.f;
  for (int c = 0; c < COUT_; ++c) mu += r[c];
  mu *= (1.f / COUT_);
  float var = 0.f;
  for (int c = 0; c < COUT_; ++c) { float d = r[c] - mu; var += d * d; }
  var *= (1.f / COUT_);
  float inv = rsqrtf(var + 1e-5f);
  int node = m / Tdim, t = m - node * Tdim;
  int b = node / Nn, n = node - b * Nn;
  for (int c = 0; c < COUT_; ++c)
    out[(((long long)b * COUT_ + c) * Tdim + t) * Nn + n] = (r[c] - mu) * inv * g[c] + bb[c];
}
__global__ void glu_ln_prop(const float* __restrict__ y2, const float* __restrict__ xa,
                            const float* __restrict__ g, const float* __restrict__ bb,
                            float* __restrict__ out, int rows, int Tdim, int Nn) {
  int m = blockIdx.x * blockDim.x + threadIdx.x;
  if (m >= rows) return;
  int node = m / Tdim, t = m - node * Tdim;
  const float* y = y2 + (long long)m * 128;
  const float* x = xa + ((long long)node * TM_ + t + 2) * COUT_;
  float p[COUT_];
  float mu = 0.f;
  for (int o = 0; o < COUT_; ++o) {
    float v = (y[o] + x[o]) * (1.f / (1.f + __expf(-y[64 + o])));
    p[o] = v; mu += v;
  }
  mu *= (1.f / COUT_);
  float var = 0.f;
  for (int o = 0; o < COUT_; ++o) { float d = p[o] - mu; var += d * d; }
  var *= (1.f / COUT_);
  float inv = rsqrtf(var + 1e-5f);
  int b = node / Nn, n = node - b * Nn;
  for (int o = 0; o < COUT_; ++o)
    out[(((long long)b * COUT_ + o) * Tdim + t) * Nn + n] = (p[o] - mu) * inv * g[o] + bb[o];
}

static inline int cdiv(long long a, int b) { return (int)((a + b - 1) / b); }

extern "C" void kernel_launch(void* const* d_in, const int* in_sizes, int n_in,
                              void* d_out, int out_size, void* d_ws, size_t ws_size,
                              hipStream_t stream) {
  (void)in_sizes; (void)n_in; (void)out_size; (void)ws_size;
  const float* x_room     = (const float*)d_in[0];
  const float* x_device   = (const float*)d_in[1];
  const float* x_property = (const float*)d_in[2];
  const float* x_outside  = (const float*)d_in[3];
  const float* x_time     = (const float*)d_in[4];
  const float* a1_room_w = (const float*)d_in[5];  const float* a1_room_b = (const float*)d_in[6];
  const float* a1_dev_w  = (const float*)d_in[7];  const float* a1_dev_b  = (const float*)d_in[8];
  const float* tc1_prop_w = (const float*)d_in[9];  const float* tc1_prop_b = (const float*)d_in[10];
  const float* tc1_out_w  = (const float*)d_in[11]; const float* tc1_out_b  = (const float*)d_in[12];
  const float* tc1_time_w = (const float*)d_in[13]; const float* tc1_time_b = (const float*)d_in[14];
  const float* g_hh_w = (const float*)d_in[15]; const float* g_hh_b = (const float*)d_in[16];
  const float* g_vv_w = (const float*)d_in[17]; const float* g_vv_b = (const float*)d_in[18];
  const float* g_dr_w = (const float*)d_in[19]; const float* g_dr_b = (const float*)d_in[20];
  const float* g_pd_w = (const float*)d_in[21]; const float* g_pd_b = (const float*)d_in[22];
  const float* g_or_w = (const float*)d_in[23]; const float* g_or_b = (const float*)d_in[24];
  const float* g_tr_w = (const float*)d_in[25]; const float* g_tr_b = (const float*)d_in[26];
  const float* g_td_w = (const float*)d_in[27]; const float* g_td_b = (const float*)d_in[28];
  const float* g_tp_w = (const float*)d_in[29]; const float* g_tp_b = (const float*)d_in[30];
  const float* a2_room_w = (const float*)d_in[31]; const float* a2_room_b = (const float*)d_in[32];
  const float* a2_dev_w  = (const float*)d_in[33]; const float* a2_dev_b  = (const float*)d_in[34];
  const float* tc2_prop_w = (const float*)d_in[35]; const float* tc2_prop_b = (const float*)d_in[36];
  const float* a2_prop_w  = (const float*)d_in[37]; const float* a2_prop_b  = (const float*)d_in[38];
  const float* ln_room_g = (const float*)d_in[39]; const float* ln_room_b = (const float*)d_in[40];
  const float* ln_dev_g  = (const float*)d_in[41]; const float* ln_dev_b  = (const float*)d_in[42];
  const float* ln_prop_g = (const float*)d_in[43]; const float* ln_prop_b = (const float*)d_in[44];
  const int* ei_hor  = (const int*)d_in[45];
  const int* ei_ver  = (const int*)d_in[46];
  const int* ei_cont = (const int*)d_in[47];
  const int* ei_meas = (const int*)d_in[48];
  const int* ei_infl = (const int*)d_in[49];
  const int* ei_tr   = (const int*)d_in[50];
  const int* ei_td   = (const int*)d_in[51];
  const int* ei_tp   = (const int*)d_in[52];

  float* ws  = (float*)d_ws;
  float* out = (float*)d_out;

  // ---- row counts ----
  constexpr int Mr = B_ * NR_ * TM_;       // 18432
  constexpr int Md = B_ * ND_ * TM_;       // 73728
  constexpr int Mp = B_ * NP_ * TM_;       // 294912
  constexpr int Mq = B_ * NP_ * (TM_ - 2); // 262144
  constexpr int Ms = B_ * TM_;             // 288

  // ---- workspace layout (floats), phase-aliased ----
  constexpr long long S_Xr = (long long)Mr * 256, S_Xd = (long long)Md * 256;
  constexpr long long S_Xp = (long long)B_ * NP_ * T_ * 32;
  constexpr long long S_Yp1 = (long long)Mp * 256;
  constexpr long long REGA = S_Xr + S_Xd + S_Xp + S_Yp1;        // phase 1-2 scratch
  constexpr long long O_Xr = 0, O_Xd = S_Xr, O_Xp = O_Xd + S_Xd, O_Yp1 = O_Xp + S_Xp;
  constexpr long long S_fr = (long long)Mr * 128, S_fd = (long long)Md * 128;
  constexpr long long S_fp = (long long)Mp * 128, S_f1 = (long long)Ms * 128;
  // phase 3 (H buffers) alias region A
  constexpr long long O_Hhh = 0, O_Hvv = S_fr, O_Hdr = 2 * S_fr, O_Hpd = O_Hdr + S_fd;
  constexpr long long O_Hor = O_Hpd + S_fp, O_Htr = O_Hor + S_f1, O_Htd = O_Htr + S_f1, O_Htp = O_Htd + S_f1;
  // phase 5 (stage-2 outputs) alias region A
  constexpr long long O_r2 = 0, O_d2 = (long long)Mr * 64, O_xa = O_d2 + (long long)Md * 64;
  constexpr long long O_yp2 = O_xa + (long long)Mp * 64;
  // region B: flat activations, later aliased by accumulators
  constexpr long long O_fr = REGA, O_fd = O_fr + S_fr, O_fp = O_fd + S_fd;
  constexpr long long O_fo = O_fp + S_fp, O_ft = O_fo + S_f1;
  constexpr long long O_racc = O_fr, O_dacc = O_fd, O_pacc = O_fp;
  // region C: weights / degree tables
  constexpr long long O_Wp1 = O_ft + S_f1, O_Wp2 = O_Wp1 + 256 * 96;
  long long cur = O_Wp2 + 128 * 384;
  float* dv_hor   = ws + cur; cur += 1024;
  float* dv_ver   = ws + cur; cur += 1024;
  float* dis_cont = ws + cur; cur += 4096;  float* did_cont = ws + cur; cur += 1024;
  float* dis_meas = ws + cur; cur += 16384; float* did_meas = ws + cur; cur += 4096;
  float* dis_infl = ws + cur; cur += 16;    float* did_infl = ws + cur; cur += 1024;
  float* dis_tr   = ws + cur; cur += 16;    float* did_tr   = ws + cur; cur += 1024;
  float* dis_td   = ws + cur; cur += 16;    float* did_td   = ws + cur; cur += 4096;
  float* dis_tp   = ws + cur; cur += 16;    float* did_tp   = ws + cur; cur += 16384;

  auto gemm = [&](const float* A, int rpn, long long cs, long long rs, int M, int K,
                  const float* W, const float* bias, float* D, int N) {
    int tilesM = M >> 4;
    int groupsN = N >> 6;
    int bm = (tilesM + 7) / 8;                       // 8 waves (M-tiles) per block
    size_t lds = (size_t)(K + 2) * 64 * sizeof(float);
    gemm_wmma_f32<<<dim3(bm * groupsN), dim3(256), lds, stream>>>(
        A, rpn, cs, rs, M, K, W, bias, D, N);
  };
  auto degs = [&](const int* ei, int E, float* dis, int ns, float* did, int nd) {
    fill_f32<<<cdiv(ns, 256), 256, 0, stream>>>(dis, ns, 0.f);
    fill_f32<<<cdiv(nd, 256), 256, 0, stream>>>(did, nd, 0.f);
    deg_count<<<cdiv(E, 256), 256, 0, stream>>>(ei, E, dis);
    deg_count<<<cdiv(E, 256), 256, 0, stream>>>(ei + E, E, did);
    to_dinv<<<cdiv(ns, 256), 256, 0, stream>>>(dis, ns);
    to_dinv<<<cdiv(nd, 256), 256, 0, stream>>>(did, nd);
  };
  auto degself = [&](const int* ei, int E, float* dv, int n) {
    fill_f32<<<cdiv(n, 256), 256, 0, stream>>>(dv, n, 1.f);   // +1 for the self loop
    deg_count<<<cdiv(E, 256), 256, 0, stream>>>(ei + E, E, dv);
    to_dinv<<<cdiv(n, 256), 256, 0, stream>>>(dv, n);
  };

  // ---- phase 1: packing ----
  pack_nchw<<<cdiv(S_Xr, 256), 256, 0, stream>>>(x_room,     ws + O_Xr, 256, T_, NR_, TM_);
  pack_nchw<<<cdiv(S_Xd, 256), 256, 0, stream>>>(x_device,   ws + O_Xd, 256, T_, ND_, TM_);
  pack_nchw<<<cdiv(S_Xp, 256), 256, 0, stream>>>(x_property, ws + O_Xp, 32,  T_, NP_, T_);
  repack_w3<<<cdiv(256 * 96, 256), 256, 0, stream>>>(tc1_prop_w, ws + O_Wp1, 256, 32);
  repack_w3<<<cdiv(128 * 384, 256), 256, 0, stream>>>(tc2_prop_w, ws + O_Wp2, 128, 128);

  // ---- phase 2: stage-1 feature transforms (WMMA GEMMs) ----
  gemm(ws + O_Xr, Mr, 0, 256, Mr, 256, a1_room_w, a1_room_b, ws + O_fr, 128);
  gemm(ws + O_Xd, Md, 0, 256, Md, 256, a1_dev_w,  a1_dev_b,  ws + O_fd, 128);
  // temporal conv as strided-view GEMM: row m=(node*18+t) at Xp + node*640 + t*32, K=96
  gemm(ws + O_Xp, TM_, (long long)T_ * 32, 32, Mp, 96, ws + O_Wp1, tc1_prop_b, ws + O_Yp1, 256);
  glu_prop1<<<cdiv((long long)Mp * 128, 256), 256, 0, stream>>>(ws + O_Yp1, ws + O_Xp, ws + O_fp);
  tconv16_glu<<<cdiv(Ms * 128, 256), 256, 0, stream>>>(x_outside, tc1_out_w,  tc1_out_b,  ws + O_fo);
  tconv16_glu<<<cdiv(Ms * 128, 256), 256, 0, stream>>>(x_time,    tc1_time_w, tc1_time_b, ws + O_ft);

  // ---- phase 3: GCN node transforms h = f @ W^T ----
  gemm(ws + O_fr, Mr, 0, 128, Mr, 128, g_hh_w, nullptr, ws + O_Hhh, 128);
  gemm(ws + O_fr, Mr, 0, 128, Mr, 128, g_vv_w, nullptr, ws + O_Hvv, 128);
  gemm(ws + O_fd, Md, 0, 128, Md, 128, g_dr_w, nullptr, ws + O_Hdr, 128);
  gemm(ws + O_fp, Mp, 0, 128, Mp, 128, g_pd_w, nullptr, ws + O_Hpd, 128);
  gemm(ws + O_fo, Ms, 0, 128, Ms, 128, g_or_w, nullptr, ws + O_Hor, 128);
  gemm(ws + O_ft, Ms, 0, 128, Ms, 128, g_tr_w, nullptr, ws + O_Htr, 128);
  gemm(ws + O_ft, Ms, 0, 128, Ms, 128, g_td_w, nullptr, ws + O_Htd, 128);
  gemm(ws + O_ft, Ms, 0, 128, Ms, 128, g_tp_w, nullptr, ws + O_Htp, 128);

  // ---- phase 4: degrees, bias-init accumulators, normalized scatters ----
  constexpr int Ehor = 3072, Ever = 1536, Econt = 4096, Emeas = 16384;
  constexpr int Einfl = 1024, Etr = 1024, Etd = 4096, Etp = 16384;
  constexpr int nRoom = B_ * NR_, nDev = B_ * ND_, nProp = B_ * NP_, nOne = B_;
  degself(ei_hor, Ehor, dv_hor, nRoom);
  degself(ei_ver, Ever, dv_ver, nRoom);
  degs(ei_cont, Econt, dis_cont, nDev,  did_cont, nRoom);
  degs(ei_meas, Emeas, dis_meas, nProp, did_meas, nDev);
  degs(ei_infl, Einfl, dis_infl, nOne,  did_infl, nRoom);
  degs(ei_tr,   Etr,   dis_tr,   nOne,  did_tr,   nRoom);
  degs(ei_td,   Etd,   dis_td,   nOne,  did_td,   nDev);
  degs(ei_tp,   Etp,   dis_tp,   nOne,  did_tp,   nProp);

  acc_init<<<cdiv((long long)Mr * 128, 256), 256, 0, stream>>>(ws + O_racc, Mr, g_hh_b, g_vv_b, g_dr_b, g_or_b, g_tr_b);
  acc_init<<<cdiv((long long)Md * 128, 256), 256, 0, stream>>>(ws + O_dacc, Md, g_pd_b, g_td_b, nullptr, nullptr, nullptr);
  acc_init<<<cdiv((long long)Mp * 128, 256), 256, 0, stream>>>(ws + O_pacc, Mp, g_tp_b, nullptr, nullptr, nullptr, nullptr);

  scatter_self <<<nRoom, 128, 0, stream>>>(ws + O_Hhh, dv_hor, ws + O_racc);
  scatter_edges<<<Ehor,  128, 0, stream>>>(ws + O_Hhh, ei_hor, Ehor, dv_hor, dv_hor, ws + O_racc);
  scatter_self <<<nRoom, 128, 0, stream>>>(ws + O_Hvv, dv_ver, ws + O_racc);
  scatter_edges<<<Ever,  128, 0, stream>>>(ws + O_Hvv, ei_ver, Ever, dv_ver, dv_ver, ws + O_racc);
  scatter_edges<<<Econt, 128, 0, stream>>>(ws + O_Hdr, ei_cont, Econt, dis_cont, did_cont, ws + O_racc);
  scatter_edges<<<Einfl, 128, 0, stream>>>(ws + O_Hor, ei_infl, Einfl, dis_infl, did_infl, ws + O_racc);
  scatter_edges<<<Etr,   128, 0, stream>>>(ws + O_Htr, ei_tr,   Etr,   dis_tr,   did_tr,   ws + O_racc);
  scatter_edges<<<Emeas, 128, 0, stream>>>(ws + O_Hpd, ei_meas, Emeas, dis_meas, did_meas, ws + O_dacc);
  scatter_edges<<<Etd,   128, 0, stream>>>(ws + O_Htd, ei_td,   Etd,   dis_td,   did_td,   ws + O_dacc);
  scatter_edges<<<Etp,   128, 0, stream>>>(ws + O_Htp, ei_tp,   Etp,   dis_tp,   did_tp,   ws + O_pacc);

  relu_inplace<<<cdiv((long long)Mr * 128, 256), 256, 0, stream>>>(ws + O_racc, (long long)Mr * 128);
  relu_inplace<<<cdiv((long long)Md * 128, 256), 256, 0, stream>>>(ws + O_dacc, (long long)Md * 128);
  relu_inplace<<<cdiv((long long)Mp * 128, 256), 256, 0, stream>>>(ws + O_pacc, (long long)Mp * 128);

  // ---- phase 5: stage-2 transforms ----
  gemm(ws + O_racc, Mr, 0, 128, Mr, 128, a2_room_w, a2_room_b, ws + O_r2, 64);
  gemm(ws + O_dacc, Md, 0, 128, Md, 128, a2_dev_w,  a2_dev_b,  ws + O_d2, 64);
  gemm(ws + O_pacc, Mp, 0, 128, Mp, 128, a2_prop_w, a2_prop_b, ws + O_xa, 64);
  // temporal conv 2: row m=(node*16+t) at pacc + node*(18*128) + t*128, K=384
  gemm(ws + O_pacc, TM_ - 2, (long long)TM_ * 128, 128, Mq, 384, ws + O_Wp2, tc2_prop_b, ws + O_yp2, 128);

  // ---- phase 6: LayerNorm epilogues -> NCHW outputs ----
  float* out_r = out;
  float* out_d = out + (long long)B_ * 64 * TM_ * NR_;
  float* out_p = out_d + (long long)B_ * 64 * TM_ * ND_;
  ln64_out<<<cdiv(Mr, 128), 128, 0, stream>>>(ws + O_r2, ln_room_g, ln_room_b, out_r, Mr, TM_, NR_);
  ln64_out<<<cdiv(Md, 128), 128, 0, stream>>>(ws + O_d2, ln_dev_g,  ln_dev_b,  out_d, Md, TM_, ND_);
  glu_ln_prop<<<cdiv(Mq, 128), 128, 0, stream>>>(ws + O_yp2, ws + O_xa, ln_prop_g, ln_prop_b,
                                                 out_p, Mq, TM_ - 2, NP_);
}